// Net_45260365365592
// MI455X (gfx1250) — compile-verified
//
#include <hip/hip_runtime.h>
#include <math.h>

#define NN  50000
#define NE  1000000
#define HID 64
#define HD2 128

typedef __attribute__((ext_vector_type(2))) float v2f;
typedef __attribute__((ext_vector_type(8))) float v8f;

// ---------------- utility ----------------
__global__ void k_zero(float* __restrict__ p, int n) {
    int i = blockIdx.x * 256 + threadIdx.x;
    if (i < n) p[i] = 0.0f;
}

// ---------------- node encoder: h = x @ node_w + node_b ----------------
__global__ void k_encode_nodes(const float* __restrict__ x, const float* __restrict__ w,
                               const float* __restrict__ b, float* __restrict__ h) {
    int i = blockIdx.x * 256 + threadIdx.x;     // over N*64
    if (i >= NN * HID) return;
    int n = i >> 6, k = i & 63;
    float acc = b[k];
#pragma unroll
    for (int j = 0; j < 3; ++j) acc += x[n * 3 + j] * w[j * HID + k];
    h[i] = acc;
}

// msg = relu(h[src]+ea)+1e-7 ; ea recomputed on the fly (edge_attr = 16MB stream,
// weights L0-resident). edge_attr read as one b128 per thread.
__device__ __forceinline__ float edge_msg(const float* __restrict__ h,
                                          const float4* __restrict__ eat4,
                                          const float* __restrict__ ew,
                                          const float* __restrict__ eb,
                                          int e, int sn, int k) {
    float4 f = eat4[e];
    float ea = eb[k] + f.x * ew[k] + f.y * ew[HID + k]
                     + f.z * ew[2 * HID + k] + f.w * ew[3 * HID + k];
    float msg = h[sn * HID + k] + ea;
    return (msg > 0.0f ? msg : 0.0f) + 1e-7f;
}

// ---------------- edge pass 1: segmented max (positive floats -> u32 atomicMax) ----------------
__global__ void k_edge_max(const float* __restrict__ h, const float4* __restrict__ eat4,
                           const int* __restrict__ src, const int* __restrict__ dst,
                           const float* __restrict__ ew, const float* __restrict__ eb,
                           float* __restrict__ m) {
    int i = blockIdx.x * 256 + threadIdx.x;     // exactly NE*HID threads
    int e = i >> 6, k = i & 63;
    float msg = edge_msg(h, eat4, ew, eb, e, src[e], k);
    atomicMax((unsigned int*)(m + dst[e] * HID + k), __float_as_uint(msg));
}

// ---------------- edge pass 2: s += exp(msg-m), t += msg*exp(msg-m) ----------------
__global__ void k_edge_sum(const float* __restrict__ h, const float4* __restrict__ eat4,
                           const int* __restrict__ src, const int* __restrict__ dst,
                           const float* __restrict__ ew, const float* __restrict__ eb,
                           const float* __restrict__ m, float* __restrict__ s,
                           float* __restrict__ t) {
    int i = blockIdx.x * 256 + threadIdx.x;
    int e = i >> 6, k = i & 63;
    int dn = dst[e];
    float msg = edge_msg(h, eat4, ew, eb, e, src[e], k);
    float ex = __expf(msg - m[dn * HID + k]);
    atomicAdd(s + dn * HID + k, ex);
    atomicAdd(t + dn * HID + k, msg * ex);
}

// ---------------- h2 = t/(s+eps) + h  (softmax-weighted aggr + residual) ----------------
__global__ void k_aggr(const float* __restrict__ h, const float* __restrict__ s,
                       const float* __restrict__ t, float* __restrict__ h2) {
    int i = blockIdx.x * 256 + threadIdx.x;
    if (i >= NN * HID) return;
    h2[i] = t[i] / (s[i] + 1e-16f) + h[i];
}

// ---------------- dense GEMM via V_WMMA_F32_16X16X4_F32 (compile-time shapes) ----------------
// out[N, NOUT] = A[N, K] @ W[K, NOUT] + bias (optional relu). One wave per 16x16 tile.
// grid.x = N/16 (N divisible by 16), blockDim.x = (NOUT/16)*32, EXEC all-1s throughout.
template <int K, int NOUT, int RELU>
__global__ void k_mlp_wmma(const float* __restrict__ A, const float* __restrict__ W,
                           const float* __restrict__ bias, float* __restrict__ out) {
    int lane = threadIdx.x & 31;
    int wv   = threadIdx.x >> 5;
    int row0 = blockIdx.x * 16;
    int col0 = wv * 16;
    // A 16x4 f32 layout: lanes 0-15 -> M=lane, K pair {0,1}; lanes 16-31 -> M=lane-16, K pair {2,3}
    int Mrow = row0 + (lane & 15);
    int koff = (lane >> 4) << 1;
    int nI   = col0 + (lane & 15);      // B/C/D column for this lane

    const float* arow = A + Mrow * K + koff;    // 8B-aligned: Mrow*K multiple of 64, koff even
    const float* wcol = W + koff * NOUT + nI;

    v8f c = {0.f, 0.f, 0.f, 0.f, 0.f, 0.f, 0.f, 0.f};
#pragma unroll
    for (int kk = 0; kk < K; kk += 4) {
        v2f a = *(const v2f*)(arow + kk);       // global_load_b64
        v2f b;
        b.x = wcol[kk * NOUT];
        b.y = wcol[(kk + 1) * NOUT];
        c = __builtin_amdgcn_wmma_f32_16x16x4_f32(false, a, false, b,
                                                  (short)0, c, false, false);
    }
    // D layout: VGPR r -> rows {r, r+8}; lanes 0-15 = first half, 16-31 = second half
    float bv = bias[nI];
    int rbase = row0 + ((lane >> 4) << 3);
#pragma unroll
    for (int r = 0; r < 8; ++r) {
        float v = c[r] + bv;
        if (RELU) v = v > 0.0f ? v : 0.0f;
        out[(rbase + r) * NOUT + nI] = v;
    }
}

// ---------------- BatchNorm stats: per-column sum and sum-of-squares ----------------
__global__ void k_bnstats(const float* __restrict__ h1, float* __restrict__ stat) {
    __shared__ float ss[256], sq[256];
    int t = threadIdx.x;
    int col = t & 127, half = t >> 7;
    int r0 = blockIdx.x * 256;
    int r1 = r0 + 256; if (r1 > NN) r1 = NN;
    float s = 0.0f, q = 0.0f;
    for (int r = r0 + half; r < r1; r += 2) {
        float v = h1[r * HD2 + col];
        s += v; q += v * v;
    }
    ss[t] = s; sq[t] = q;
    __syncthreads();
    if (half == 0) {
        atomicAdd(&stat[col],       ss[t] + ss[t + 128]);
        atomicAdd(&stat[128 + col], sq[t] + sq[t + 128]);
    }
}

// ---------------- fold BN into scale/shift ----------------
__global__ void k_bnfin(float* __restrict__ stat, const float* __restrict__ g,
                        const float* __restrict__ bt) {
    int c = threadIdx.x;                 // 128 threads
    float mu  = stat[c] * (1.0f / NN);
    float var = stat[128 + c] * (1.0f / NN) - mu * mu;
    float sc  = g[c] * rsqrtf(var + 1e-5f);
    stat[256 + c] = sc;
    stat[384 + c] = bt[c] - mu * sc;
}

// ---------------- h1 = relu(h1*scale + shift) in place ----------------
__global__ void k_bnrelu(float* __restrict__ h1, const float* __restrict__ stat) {
    int i = blockIdx.x * 256 + threadIdx.x;
    if (i >= NN * HD2) return;
    int col = i & 127;
    float v = h1[i] * stat[256 + col] + stat[384 + col];
    h1[i] = v > 0.0f ? v : 0.0f;
}

// ---------------- final layer: [128]->1 dot + sigmoid, one wave32 per node ----------------
__global__ void k_final(const float* __restrict__ hn, const float* __restrict__ w,
                        const float* __restrict__ b, float* __restrict__ out) {
    int lane = threadIdx.x & 31;
    int node = blockIdx.x * 8 + (threadIdx.x >> 5);
    if (node >= NN) return;
    float v = 0.0f;
#pragma unroll
    for (int j = 0; j < 4; ++j)
        v += hn[node * HD2 + lane + j * 32] * w[lane + j * 32];
    for (int off = 16; off > 0; off >>= 1) v += __shfl_down(v, off, 32);
    if (lane == 0) out[node] = 1.0f / (1.0f + __expf(-(v + b[0])));
}

extern "C" void kernel_launch(void* const* d_in, const int* in_sizes, int n_in,
                              void* d_out, int out_size, void* d_ws, size_t ws_size,
                              hipStream_t stream) {
    const float*  x    = (const float*) d_in[0];
    const float4* eat4 = (const float4*)d_in[1];
    const int*    ei   = (const int*)   d_in[2];
    const float*  nw   = (const float*) d_in[3];
    const float*  nb   = (const float*) d_in[4];
    const float*  ew   = (const float*) d_in[5];
    const float*  eb   = (const float*) d_in[6];
    const float*  cw1  = (const float*) d_in[7];
    const float*  cb1  = (const float*) d_in[8];
    const float*  cg   = (const float*) d_in[9];
    const float*  cbt  = (const float*) d_in[10];
    const float*  cw2  = (const float*) d_in[11];
    const float*  cb2  = (const float*) d_in[12];
    const float*  c4w1 = (const float*) d_in[13];
    const float*  c4b1 = (const float*) d_in[14];
    const float*  c4g  = (const float*) d_in[15];
    const float*  c4bt = (const float*) d_in[16];
    const float*  c4w2 = (const float*) d_in[17];
    const float*  c4b2 = (const float*) d_in[18];
    const int* src = ei;          // edge_index[0]
    const int* dst = ei + NE;     // edge_index[1]

    float* ws   = (float*)d_ws;
    float* h    = ws;                             // [N,64]
    float* h2   = ws + (size_t)NN * HID;          // [N,64]
    float* m    = ws + 2 * (size_t)NN * HID;      // [N,64]
    float* s    = ws + 3 * (size_t)NN * HID;      // [N,64]
    float* t    = ws + 4 * (size_t)NN * HID;      // [N,64]
    float* h1   = m;                              // [N,128] reuses m+s regions (MLP phase)
    float* stat = ws + 5 * (size_t)NN * HID;      // 512 floats

    k_encode_nodes<<<(NN * HID + 255) / 256, 256, 0, stream>>>(x, nw, nb, h);

    const int EG = (NE * HID) / 256;              // 250000 blocks, exact
    for (int L = 0; L < 4; ++L) {
        const float* w1 = (L < 3) ? cw1 + (size_t)L * HID * HD2 : c4w1;
        const float* b1 = (L < 3) ? cb1 + L * HD2 : c4b1;
        const float* g  = (L < 3) ? cg  + L * HD2 : c4g;
        const float* bt = (L < 3) ? cbt + L * HD2 : c4bt;
        const float* w2 = (L < 3) ? cw2 + (size_t)L * HD2 * HID : c4w2;
        const float* b2 = (L < 3) ? cb2 + L * HID : c4b2;

        k_zero<<<(3 * NN * HID + 255) / 256, 256, 0, stream>>>(m, 3 * NN * HID);
        k_zero<<<2, 256, 0, stream>>>(stat, 512);
        k_edge_max<<<EG, 256, 0, stream>>>(h, eat4, src, dst, ew, eb, m);
        k_edge_sum<<<EG, 256, 0, stream>>>(h, eat4, src, dst, ew, eb, m, s, t);
        k_aggr<<<(NN * HID + 255) / 256, 256, 0, stream>>>(h, s, t, h2);

        k_mlp_wmma<HID, HD2, 0><<<NN / 16, 256, 0, stream>>>(h2, w1, b1, h1);
        k_bnstats<<<(NN + 255) / 256, 256, 0, stream>>>(h1, stat);
        k_bnfin<<<1, 128, 0, stream>>>(stat, g, bt);
        k_bnrelu<<<(NN * HD2 + 255) / 256, 256, 0, stream>>>(h1, stat);

        if (L < 3) {
            k_mlp_wmma<HD2, HID, 1><<<NN / 16, 128, 0, stream>>>(h1, w2, b2, h);
        } else {
            k_final<<<(NN + 7) / 8, 256, 0, stream>>>(h1, w2, b2, (float*)d_out);
        }
    }
}